// InterRankAttention_69329362092306
// MI455X (gfx1250) — compile-verified
//
#include <hip/hip_runtime.h>
#include <hip/hip_bf16.h>
#include <math.h>

// Problem constants (match reference)
#define NSRC  50000
#define NTGT  50000
#define NEDGE 400000
#define EMB   128
#define NHID  512   // HEADS * HID = 4 * 128

typedef __attribute__((ext_vector_type(2))) float v2f;
typedef __attribute__((ext_vector_type(8))) float v8f;

// ---- order-preserving float <-> uint key for atomic max -------------------
__device__ __forceinline__ unsigned fkey(float f) {
    unsigned u = __float_as_uint(f);
    return (u & 0x80000000u) ? ~u : (u | 0x80000000u);
}
__device__ __forceinline__ float fdec(unsigned k) {
    unsigned u = (k & 0x80000000u) ? (k & 0x7FFFFFFFu) : ~k;
    return __uint_as_float(u);
}

// ---- workspace init (agg=0, den=0, mx=key(-inf)=0) ------------------------
__global__ void __launch_bounds__(256) init_ws_kernel(float* __restrict__ agg,
                                                      float* __restrict__ den,
                                                      unsigned* __restrict__ mx) {
    int i = blockIdx.x * blockDim.x + threadIdx.x;
    if (i < NTGT * EMB) agg[i] = 0.0f;
    if (i < NTGT * 4) { den[i] = 0.0f; mx[i] = 0u; }
}

// ---- fp32 WMMA GEMM: C[M,N] = A[M,128] @ W[128,N] + bias ------------------
// One wave computes a 16x64 output tile: 4 accumulators, K stepped by 4 using
// v_wmma_f32_16x16x4_f32. All five fragments of a k-step are fetched before
// the four back-to-back WMMAs, so one loadcnt wait covers the whole group and
// the unrolled loop can software-pipeline loads over the matrix ops.
// Requires M % 16 == 0, N % 64 == 0 (true for all four GEMMs here).
__global__ void __launch_bounds__(256) gemm_k128_wmma(
    const float* __restrict__ A, const float* __restrict__ W,
    const float* __restrict__ bias, float* __restrict__ C,
    int Mrows, int Ncols)
{
    const int K = 128;
    int wave = (blockIdx.x * blockDim.x + threadIdx.x) >> 5;
    int lane = threadIdx.x & 31;
    int nt64 = Ncols >> 6;
    int total = (Mrows >> 4) * nt64;
    if (wave >= total) return;

    int tm = wave / nt64;
    int tn = wave - tm * nt64;
    int kb = (lane >> 4) * 2;                 // half-wave K split (K / K+2)
    const float* Arow = A + (size_t)(tm * 16 + (lane & 15)) * K;
    int ncb = tn * 64 + (lane & 15);

    v8f acc[4] = {};                          // zero-init accumulators
#pragma unroll 4
    for (int k = 0; k < K; k += 4) {
        // Fetch A fragment (contiguous pair -> b64 load)
        v2f a;
        a.x = Arow[k + kb];
        a.y = Arow[k + kb + 1];
        // Fetch all four B fragments for this k-step first
        const float* W0 = W + (size_t)(k + kb) * Ncols + ncb;
        const float* W1 = W0 + Ncols;
        v2f b[4];
#pragma unroll
        for (int nt = 0; nt < 4; ++nt) {
            b[nt].x = W0[nt * 16];
            b[nt].y = W1[nt * 16];
        }
        // Then issue the four independent WMMAs back to back
#pragma unroll
        for (int nt = 0; nt < 4; ++nt) {
            acc[nt] = __builtin_amdgcn_wmma_f32_16x16x4_f32(
                false, a, false, b[nt], (short)0, acc[nt], false, false);
        }
    }

    // C/D layout: lanes 0-15 -> M=r, lanes 16-31 -> M=r+8; N = lane&15
    int rowb = tm * 16 + ((lane >> 4) * 8);
#pragma unroll
    for (int nt = 0; nt < 4; ++nt) {
        int col = ncb + nt * 16;
        float bv = bias[col];
#pragma unroll
        for (int r = 0; r < 8; ++r) {
            C[(size_t)(rowb + r) * Ncols + col] = acc[nt][r] + bv;
        }
    }
}

// ---- per-edge logits + segment max ----------------------------------------
// One wave per edge. Lane l reads elements l, l+32, ..., l+480 (coalesced).
// Element index idx = 32k + l lies entirely in head (k>>2), so per-lane
// partials accumulate cleanly; butterfly-shuffle reduces 32 lanes per head.
__global__ void __launch_bounds__(256) edge_logits_kernel(
    const float* __restrict__ Hs, const float* __restrict__ Ht,
    const int* __restrict__ esrc, const int* __restrict__ etgt,
    const float* __restrict__ attn_w,
    float* __restrict__ lg, unsigned* __restrict__ mx)
{
    int e = (blockIdx.x * blockDim.x + threadIdx.x) >> 5;
    int lane = threadIdx.x & 31;
    if (e >= NEDGE) return;
    int s = esrc[e], t = etgt[e];
    const float* ps = Hs + (size_t)s * NHID;
    const float* pt = Ht + (size_t)t * NHID;

    float hsum[4] = {0.f, 0.f, 0.f, 0.f};
#pragma unroll
    for (int k = 0; k < 16; ++k) {
        int idx = k * 32 + lane;
        float v = ps[idx] + pt[idx];
        // exact GELU: 0.5*v*(1+erf(v/sqrt(2)))
        float g = 0.5f * v * (1.0f + erff(v * 0.70710678118654752f));
        hsum[k >> 2] += g * attn_w[idx];      // attn_w flat [4,128] == idx
    }
#pragma unroll
    for (int h = 0; h < 4; ++h) {
#pragma unroll
        for (int off = 16; off; off >>= 1)
            hsum[h] += __shfl_xor(hsum[h], off, 32);
    }
    if (lane < 4) {
        float L = hsum[lane];
        lg[(size_t)e * 4 + lane] = L;
        atomicMax(&mx[t * 4 + lane], fkey(L));
    }
}

// ---- exp(logit - mx) + segment sum ----------------------------------------
__global__ void __launch_bounds__(256) edge_exp_kernel(
    const int* __restrict__ etgt, float* __restrict__ lg,
    const unsigned* __restrict__ mx, float* __restrict__ den)
{
    int i = blockIdx.x * blockDim.x + threadIdx.x;
    if (i >= NEDGE * 4) return;
    int e = i >> 2, h = i & 3;
    int t = etgt[e];
    float m = fdec(mx[t * 4 + h]);
    float ex = __expf(lg[i] - m);
    lg[i] = ex;                               // in-place: logit -> ex
    atomicAdd(&den[t * 4 + h], ex);
}

// ---- weighted message scatter ---------------------------------------------
// One wave per edge; iteration k covers head k (elements 32k..32k+31).
__global__ void __launch_bounds__(256) edge_scatter_kernel(
    const int* __restrict__ esrc, const int* __restrict__ etgt,
    const float* __restrict__ Hm, const float* __restrict__ lg,
    const float* __restrict__ den, float* __restrict__ agg)
{
    int e = (blockIdx.x * blockDim.x + threadIdx.x) >> 5;
    int lane = threadIdx.x & 31;
    if (e >= NEDGE) return;
    int s = esrc[e], t = etgt[e];
    const float* pm = Hm + (size_t)s * EMB;
    float* pa = agg + (size_t)t * EMB;
#pragma unroll
    for (int k = 0; k < 4; ++k) {
        float sc = lg[(size_t)e * 4 + k] / den[t * 4 + k];
        int j = k * 32 + lane;
        atomicAdd(&pa[j], pm[j] * sc);
    }
}

extern "C" void kernel_launch(void* const* d_in, const int* in_sizes, int n_in,
                              void* d_out, int out_size, void* d_ws, size_t ws_size,
                              hipStream_t stream) {
    const float* src_f  = (const float*)d_in[0];
    const float* tgt_f  = (const float*)d_in[1];
    const int*   etgt   = (const int*)d_in[2];
    const int*   esrc   = (const int*)d_in[3];
    const float* Ws     = (const float*)d_in[4];
    const float* bs     = (const float*)d_in[5];
    const float* Wt     = (const float*)d_in[6];
    const float* bt     = (const float*)d_in[7];
    const float* attn_w = (const float*)d_in[8];
    const float* Wm     = (const float*)d_in[9];
    const float* bm     = (const float*)d_in[10];
    const float* Wo     = (const float*)d_in[11];
    const float* bo     = (const float*)d_in[12];
    float* out = (float*)d_out;

    // Workspace carve-up (~264 MB)
    char* p = (char*)d_ws;
    float* Hs  = (float*)p; p += (size_t)NSRC * NHID * sizeof(float);   // 102.4 MB
    float* Ht  = (float*)p; p += (size_t)NTGT * NHID * sizeof(float);   // 102.4 MB
    float* Hm  = (float*)p; p += (size_t)NSRC * EMB  * sizeof(float);   //  25.6 MB
    float* agg = (float*)p; p += (size_t)NTGT * EMB  * sizeof(float);   //  25.6 MB
    float* lg  = (float*)p; p += (size_t)NEDGE * 4   * sizeof(float);   //   6.4 MB
    float* den = (float*)p; p += (size_t)NTGT * 4    * sizeof(float);   //   0.8 MB
    unsigned* mx = (unsigned*)p;                                        //   0.8 MB

    init_ws_kernel<<<(NTGT * EMB + 255) / 256, 256, 0, stream>>>(agg, den, mx);

    {   // Hs = src @ Ws + bs   [50000 x 512]
        int waves = (NSRC / 16) * (NHID / 64);
        gemm_k128_wmma<<<(waves * 32 + 255) / 256, 256, 0, stream>>>(
            src_f, Ws, bs, Hs, NSRC, NHID);
    }
    {   // Ht = tgt @ Wt + bt   [50000 x 512]
        int waves = (NTGT / 16) * (NHID / 64);
        gemm_k128_wmma<<<(waves * 32 + 255) / 256, 256, 0, stream>>>(
            tgt_f, Wt, bt, Ht, NTGT, NHID);
    }
    {   // Hm = src @ Wm + bm   [50000 x 128]
        int waves = (NSRC / 16) * (EMB / 64);
        gemm_k128_wmma<<<(waves * 32 + 255) / 256, 256, 0, stream>>>(
            src_f, Wm, bm, Hm, NSRC, EMB);
    }

    edge_logits_kernel<<<(NEDGE * 32 + 255) / 256, 256, 0, stream>>>(
        Hs, Ht, esrc, etgt, attn_w, lg, mx);
    edge_exp_kernel<<<(NEDGE * 4 + 255) / 256, 256, 0, stream>>>(
        etgt, lg, mx, den);
    edge_scatter_kernel<<<(NEDGE * 32 + 255) / 256, 256, 0, stream>>>(
        esrc, etgt, Hm, lg, den, agg);

    {   // out = agg @ Wo + bo  [50000 x 128]
        int waves = (NTGT / 16) * (EMB / 64);
        gemm_k128_wmma<<<(waves * 32 + 255) / 256, 256, 0, stream>>>(
            agg, Wo, bo, out, NTGT, EMB);
    }
}